// LeakyCascade_30485677868019
// MI455X (gfx1250) — compile-verified
//
#include <hip/hip_runtime.h>

#define NB 8
#define NT 4096
#define ND 256
#define NM 8
#define NCH (NT / 16)

typedef __attribute__((ext_vector_type(2))) float v2f;
typedef __attribute__((ext_vector_type(8))) float v8f;
typedef __attribute__((ext_vector_type(4))) int v4i;

typedef __attribute__((address_space(1))) v4i* gbl_v4i_p;
typedef __attribute__((address_space(3))) v4i* lds_v4i_p;

#if defined(__HIP_DEVICE_COMPILE__)
#if __has_builtin(__builtin_amdgcn_wmma_f32_16x16x4_f32)
#define USE_WMMA 1
#endif
#if __has_builtin(__builtin_amdgcn_global_load_async_to_lds_b128) && \
    __has_builtin(__builtin_amdgcn_s_wait_asynccnt)
#define USE_ASYNC 1
#endif
#endif

// Kernel 0: alphaT[(b*NM+m)*NT + t] = exp(-dt[b,t]/tau[m]),
// dt[b,0]=0, dt[b,t]=max(delta[b,t]-delta[b,t-1],0). Transposed layout so the
// scan kernel's per-chunk alpha load (16 consecutive t) is coalesced.
__global__ void lc_alpha_kernel(const float* __restrict__ delta,
                                const float* __restrict__ tau,
                                float* __restrict__ alphaT) {
  int idx = blockIdx.x * blockDim.x + threadIdx.x;
  if (idx >= NB * NT) return;
  int b = idx / NT, t = idx - b * NT;
  float dt = 0.0f;
  if (t > 0) dt = fmaxf(delta[b * NT + t] - delta[b * NT + t - 1], 0.0f);
#pragma unroll
  for (int m = 0; m < NM; ++m) {
    alphaT[((size_t)(b * NM + m)) * NT + t] = __expf(-dt / tau[m]);
  }
}

// Scan kernel: block = (b, d-tile of 16), 8 waves = 8 values of m.
// Each chunk of 16 timesteps: out_tile(16x16) = L(16x16) @ x_tile(16x16) + c ⊗ s0
// computed as 4 chained V_WMMA_F32_16X16X4_F32. x_tile staged to LDS once per
// block (shared by all 8 m-waves) via async global->LDS, double buffered.
__launch_bounds__(256) __global__
void lc_scan_kernel(const float* __restrict__ x,
                    const float* __restrict__ alphaT,
                    float* __restrict__ out) {
  const int tid  = threadIdx.x;
  const int wave = tid >> 5;
  const int lane = tid & 31;
  const int li   = lane & 15;
  const int half = lane >> 4;
  const int b    = blockIdx.x >> 4;
  const int d0   = (blockIdx.x & 15) * 16;
  const int m    = wave;

  __shared__ __align__(16) float xs[2][16][16];

  const float* xbase = x + (size_t)b * NT * ND + d0;
  const float* aT    = alphaT + (size_t)(b * NM + m) * NT;
  float* obase       = out + ((size_t)b * NT) * (NM * ND) + m * ND + d0 + li;

  // Stage chunk c's 16x16 x-slice into xs[buf]. Wave 0 only: 32 lanes cover
  // 16 rows x 64B in two 512B B128 transfers.
  auto stage = [&](int buf, int c) {
#pragma unroll
    for (int k = 0; k < 2; ++k) {
      int row = k * 8 + (lane >> 2);
      int col = (lane & 3) * 4;
      const float* g = xbase + (size_t)(c * 16 + row) * ND + col;
      float* l = &xs[buf][row][col];
#if defined(USE_ASYNC)
      __builtin_amdgcn_global_load_async_to_lds_b128(
          (gbl_v4i_p)const_cast<float*>(g), (lds_v4i_p)l, 0, 0);
#else
      *(float4*)l = *(const float4*)g;
#endif
    }
  };

  if (wave == 0) {
    stage(0, 0);
    stage(1, 1);
  }

  float s0 = 0.0f;  // carry state for d = d0 + li (replicated in both halves)

  for (int c = 0; c < NCH; ++c) {
#if defined(USE_ASYNC)
    if (wave == 0) {
      if (c < NCH - 1) __builtin_amdgcn_s_wait_asynccnt(2);  // chunk c landed
      else             __builtin_amdgcn_s_wait_asynccnt(0);
    }
#endif
    __syncthreads();  // staged data visible to all 8 waves

    const int t0  = c * 16;
    const int buf = c & 1;
    float a = aT[t0 + li];  // a_{li} of this chunk (dup in upper half)

#if defined(USE_WMMA)
    // Row li of L: L[i][j] = (1-a_j) * prod_{k=j+1..i} a_k (j<=i), and
    // c_i = prod_{k=0..i} a_k. Built right-to-left with lane broadcasts;
    // all register indices static.
    float Lr[16];
    float P = 1.0f;
#pragma unroll
    for (int j = 15; j >= 0; --j) {
      float aj = __uint_as_float(__builtin_amdgcn_readlane(__float_as_uint(a), j));
      bool act = (li >= j);
      Lr[j] = act ? (1.0f - aj) * P : 0.0f;
      P = act ? P * aj : P;
    }
    const float ci = P;  // c_{li}

    // Accumulator C[r] holds out[r + 8*half][d0+li]; init with c_i * s0.
    v8f Cv;
#pragma unroll
    for (int r = 0; r < 8; ++r) {
      float clo = __uint_as_float(__builtin_amdgcn_readlane(__float_as_uint(ci), r));
      float chi = __uint_as_float(__builtin_amdgcn_readlane(__float_as_uint(ci), r + 8));
      Cv[r] = (half ? chi : clo) * s0;
    }

    // out_tile += L @ x_tile via 4 chained K=4 WMMAs.
    // NOTE: pull the four candidate L elements into named scalars with
    // constant indices BEFORE the half-select, so clang emits one v_cndmask
    // per operand instead of folding to a dynamic array index (which SROA
    // lowers as a 15-deep cmp/cndmask select tree).
#pragma unroll
    for (int q = 0; q < 4; ++q) {
      const float l0 = Lr[4 * q + 0];
      const float l1 = Lr[4 * q + 1];
      const float l2 = Lr[4 * q + 2];
      const float l3 = Lr[4 * q + 3];
      v2f A, Bv;
      A.x = half ? l2 : l0;  // A vgpr0: K0 (lanes 0-15) / K2 (lanes 16-31)
      A.y = half ? l3 : l1;  // A vgpr1: K1 / K3
      const float* xr = &xs[buf][4 * q + 2 * half][0];
      Bv.x = xr[li];         // B vgpr0: K0 / K2
      Bv.y = xr[16 + li];    // B vgpr1: K1 / K3  (next row, +16 floats)
      Cv = __builtin_amdgcn_wmma_f32_16x16x4_f32(false, A, false, Bv,
                                                 (short)0, Cv, false, false);
    }

    // Streaming stores (output is write-once; keep it out of L2).
#pragma unroll
    for (int r = 0; r < 8; ++r) {
      __builtin_nontemporal_store(Cv[r],
          obase + (size_t)(t0 + r + 8 * half) * (NM * ND));
    }
    // New carry = out row 15 = Cv[7] in lanes 16..31; broadcast to all lanes.
    s0 = __uint_as_float(
        __builtin_amdgcn_ds_bpermute((16 + li) << 2, __float_as_uint(Cv[7])));
#else
    // Scalar fallback (correct, no WMMA): per-lane recurrence within chunk.
    float s = s0;
#pragma unroll
    for (int i = 0; i < 16; ++i) {
      float ai = __uint_as_float(__builtin_amdgcn_readlane(__float_as_uint(a), i));
      s = ai * s + (1.0f - ai) * xs[buf][i][li];
      __builtin_nontemporal_store(s, obase + (size_t)(t0 + i) * (NM * ND));
    }
    s0 = s;
#endif

    __syncthreads();  // everyone done reading xs[buf] before it is re-staged
    if (wave == 0 && c + 2 < NCH) stage(buf, c + 2);
  }
}

extern "C" void kernel_launch(void* const* d_in, const int* in_sizes, int n_in,
                              void* d_out, int out_size, void* d_ws,
                              size_t ws_size, hipStream_t stream) {
  const float* x     = (const float*)d_in[0];
  const float* delta = (const float*)d_in[1];
  const float* tau   = (const float*)d_in[2];
  float* out         = (float*)d_out;
  float* alphaT      = (float*)d_ws;  // NB*NM*NT floats = 1 MB

  (void)in_sizes; (void)n_in; (void)out_size; (void)ws_size;

  lc_alpha_kernel<<<(NB * NT + 255) / 256, 256, 0, stream>>>(delta, tau, alphaT);
  lc_scan_kernel<<<NB * (ND / 16), 256, 0, stream>>>(x, alphaT, out);
}